// Node_AE_83949430768185
// MI455X (gfx1250) — compile-verified
//
#include <hip/hip_runtime.h>

#define N_NODES   10000
#define N_EDGES   320000
#define IN_NODE   11
#define IN_EDGE   4
#define H_NF      4
#define OUT_NF    4
#define EMB_NF    2

typedef float v2f __attribute__((ext_vector_type(2)));
typedef float v8f __attribute__((ext_vector_type(8)));

// ---------------- Kernel 1: zero the aggregation buffer ----------------
__global__ void zero_agg_kernel(float* __restrict__ agg) {
    int t = blockIdx.x * blockDim.x + threadIdx.x;
    if (t < N_NODES * IN_EDGE) agg[t] = 0.0f;
}

// ---------------- Kernel 2: scatter-add edge_attr into agg ----------------
// one thread per (edge, feature) -> coalesced attr loads, float atomics into
// a 160KB L2-resident buffer.
__global__ void edge_scatter_kernel(const float* __restrict__ edge_attr,
                                    const int*   __restrict__ edge_index, // row 0 = dst
                                    float*       __restrict__ agg) {
    int t = blockIdx.x * blockDim.x + threadIdx.x;
    if (t >= N_EDGES * IN_EDGE) return;
    int e = t >> 2;
    int k = t & 3;
    float v = edge_attr[t];
    int dst = edge_index[e];                 // edge_index[0][e]
    atomicAdd(&agg[dst * IN_EDGE + k], v);
}

// ---------------- Kernel 3: per-node MLP -> emb, norms ----------------
__global__ void mlp_kernel(const float* __restrict__ node_feats,
                           const float* __restrict__ agg,
                           const float* __restrict__ W1, const float* __restrict__ b1,
                           const float* __restrict__ W2, const float* __restrict__ b2,
                           const float* __restrict__ We, const float* __restrict__ be,
                           float* __restrict__ emb_out,   // d_out[0 : 2N)
                           float* __restrict__ emb_ws,    // ws copy for decode
                           float* __restrict__ nrm) {     // ||e_i||^2
    int i = blockIdx.x * blockDim.x + threadIdx.x;
    if (i >= N_NODES) return;

    float x[IN_NODE + IN_EDGE];
    #pragma unroll
    for (int k = 0; k < IN_NODE; ++k) x[k] = node_feats[i * IN_NODE + k];
    #pragma unroll
    for (int k = 0; k < IN_EDGE; ++k) x[IN_NODE + k] = agg[i * IN_EDGE + k];

    float h[H_NF];
    #pragma unroll
    for (int j = 0; j < H_NF; ++j) {
        float s = b1[j];
        #pragma unroll
        for (int k = 0; k < IN_NODE + IN_EDGE; ++k)
            s = fmaf(x[k], W1[k * H_NF + j], s);
        h[j] = fmaxf(s, 0.0f);
    }
    float o[OUT_NF];
    #pragma unroll
    for (int j = 0; j < OUT_NF; ++j) {
        float s = b2[j];
        #pragma unroll
        for (int k = 0; k < H_NF; ++k)
            s = fmaf(h[k], W2[k * OUT_NF + j], s);
        o[j] = s;
    }
    float e0 = be[0], e1 = be[1];
    #pragma unroll
    for (int k = 0; k < OUT_NF; ++k) {
        e0 = fmaf(o[k], We[k * EMB_NF + 0], e0);
        e1 = fmaf(o[k], We[k * EMB_NF + 1], e1);
    }
    emb_out[2 * i + 0] = e0;
    emb_out[2 * i + 1] = e1;
    emb_ws[2 * i + 0]  = e0;
    emb_ws[2 * i + 1]  = e1;
    nrm[i] = e0 * e0 + e1 * e1;
}

// ---------------- Kernel 4: pairwise decode via WMMA ----------------
// One wave per 16x16 tile of adj. Gram cross-term E @ E^T (K=2 padded to 4)
// computed with one v_wmma_f32_16x16x4_f32 per tile.
//
// sigmoid(3*d2 - 1) with d2 = ni + nj - 2*dot is algebraically folded into
// the exp2 argument:
//   arg = C1*ni + C1*nj - 2*C1*dot + C2 ,  C1 = -3*log2(e), C2 = log2(e)
//   adj = rcp(1 + exp2(arg))
// so per element: 1 fma + exp2 + add + rcp + select + store (row offsets are
// compile-time immediates off a single 64-bit base pointer).
__global__ void __launch_bounds__(256)
decode_kernel(const float* __restrict__ emb,
              const float* __restrict__ nrm,
              float* __restrict__ adj) {
    const int NT = N_NODES / 16;                 // 625, exact
    const int total_tiles = NT * NT;             // 390625
    int lane = threadIdx.x & 31;
    int w = (blockIdx.x * blockDim.x + threadIdx.x) >> 5;   // global wave id
    if (w >= total_tiles) return;                // wave-uniform: EXEC all-1s inside

    int row_tile = w / NT;
    int col_tile = w - row_tile * NT;
    int row_base = row_tile << 4;
    int col_base = col_tile << 4;
    int half = lane >> 4;                        // 0: K=0,1 ; 1: K=2,3 (zero pad)
    int l16  = lane & 15;

    // A: 16x4 (rows = lane%16, VGPR = K, lane-half selects K pair)
    // B: 4x16 (cols = lane%16, same K mapping)
    v2f a = {0.0f, 0.0f};
    v2f b = {0.0f, 0.0f};
    if (half == 0) {
        a.x = emb[2 * (row_base + l16) + 0];     // K=0 : e.x
        a.y = emb[2 * (row_base + l16) + 1];     // K=1 : e.y
        b.x = emb[2 * (col_base + l16) + 0];
        b.y = emb[2 * (col_base + l16) + 1];
    }
    v8f c = {};
    c = __builtin_amdgcn_wmma_f32_16x16x4_f32(false, a, false, b,
                                              (short)0, c, false, false);

    const float LOG2E = 1.4426950408889634f;
    const float C1 = -3.0f * LOG2E;              // DEC_W folded
    const float C2 = LOG2E;                      // -DEC_B folded
    const float CD = -2.0f * C1;                 // coefficient on dot

    // norms: nj per lane (its column); ni for the 8 rows this lane-half owns
    float nj = nrm[col_base + l16];
    const float4* np = (const float4*)&nrm[row_base + half * 8];
    float4 n0 = np[0];
    float4 n1 = np[1];
    float ni[8] = {n0.x, n0.y, n0.z, n0.w, n1.x, n1.y, n1.z, n1.w};

    float tj = fmaf(C1, nj, C2);
    float ci[8];
    #pragma unroll
    for (int v = 0; v < 8; ++v) ci[v] = fmaf(C1, ni[v], tj);

    int col  = col_base + l16;
    int row0 = row_base + half * 8;              // C layout: M = v + 8*half
    // single 64-bit base; per-row offsets become 24-bit instruction immediates
    float* __restrict__ p = adj + ((long long)row0 * N_NODES + col);

    #pragma unroll
    for (int v = 0; v < 8; ++v) {
        float arg = fmaf(CD, c[v], ci[v]);
        float ev  = __builtin_amdgcn_exp2f(arg);
        float av  = __builtin_amdgcn_rcpf(1.0f + ev);
        if (row0 + v == col) av = 0.0f;          // zero diagonal
        __builtin_nontemporal_store(av, p + v * N_NODES);
    }
}

extern "C" void kernel_launch(void* const* d_in, const int* in_sizes, int n_in,
                              void* d_out, int out_size, void* d_ws, size_t ws_size,
                              hipStream_t stream) {
    const float* node_feats = (const float*)d_in[0];
    const int*   edge_index = (const int*)  d_in[1];   // [2, E]; row 0 = dst
    const float* edge_attr  = (const float*)d_in[2];
    const float* W1  = (const float*)d_in[3];
    const float* b1  = (const float*)d_in[4];
    const float* W2  = (const float*)d_in[5];
    const float* b2  = (const float*)d_in[6];
    const float* We  = (const float*)d_in[7];
    const float* be  = (const float*)d_in[8];

    float* out = (float*)d_out;
    float* emb_out = out;                        // [N, 2]
    float* adj     = out + 2 * N_NODES;          // [N, N]

    float* ws     = (float*)d_ws;
    float* agg    = ws;                          // 40000 floats
    float* emb_ws = ws + N_NODES * IN_EDGE;      // 20000 floats
    float* nrm    = emb_ws + N_NODES * EMB_NF;   // 10000 floats

    zero_agg_kernel<<<(N_NODES * IN_EDGE + 255) / 256, 256, 0, stream>>>(agg);

    edge_scatter_kernel<<<(N_EDGES * IN_EDGE + 255) / 256, 256, 0, stream>>>(
        edge_attr, edge_index, agg);

    mlp_kernel<<<(N_NODES + 255) / 256, 256, 0, stream>>>(
        node_feats, agg, W1, b1, W2, b2, We, be, emb_out, emb_ws, nrm);

    const int NT = N_NODES / 16;                 // 625
    const int total_tiles = NT * NT;             // 390625 tiles = waves
    const int waves_per_block = 256 / 32;        // 8
    const int blocks = (total_tiles + waves_per_block - 1) / waves_per_block;
    decode_kernel<<<blocks, 256, 0, stream>>>(emb_ws, nrm, adj);
}